// SSIMError_42803644072670
// MI455X (gfx1250) — compile-verified
//
#include <hip/hip_runtime.h>
#include <cstdint>
#include <cstddef>

#ifndef __has_builtin
#define __has_builtin(x) 0
#endif

// CDNA5 async global->LDS copies (ASYNCcnt path). Fall back to plain
// load/store staging if this toolchain doesn't expose the builtins.
#if defined(__AMDGCN__) && __has_builtin(__builtin_amdgcn_global_load_async_to_lds_b32) && \
    __has_builtin(__builtin_amdgcn_s_wait_asynccnt)
#define USE_ASYNC_LDS 1
#else
#define USE_ASYNC_LDS 0
#endif

typedef __attribute__((address_space(1))) int g_int;   // global (device) dword
typedef __attribute__((address_space(3))) int l_int;   // LDS dword

static constexpr int H  = 384;
static constexpr int W  = 640;
static constexpr int TW = 64;        // tile width  (output)
static constexpr int TH = 16;        // tile height (output)
static constexpr int LW = TW + 2;    // 66, staged width incl. halo
static constexpr int LH = TH + 2;    // 18, staged height incl. halo

static constexpr float EPS1 = 1.0e-4f;   // 0.01^2
static constexpr float EPS2 = 9.0e-4f;   // 0.03^2

__global__ __launch_bounds__(256) void ssim_err_kernel(const float* __restrict__ pred,
                                                       const float* __restrict__ targ,
                                                       float* __restrict__ out) {
  __shared__ float xs[LH * LW];
  __shared__ float ys[LH * LW];

  const int tx = threadIdx.x;              // 0..63  -> output column in tile
  const int ty = threadIdx.y;              // 0..3   -> 4-row strip in tile
  const int bx = blockIdx.x;
  const int by = blockIdx.y;
  const size_t pbase = (size_t)blockIdx.z * (size_t)(H * W);  // plane base

  // ---------------- Stage (TH+2)x(TW+2) halo tiles of pred/targ into LDS ----
  // Reflect-101 padding resolved arithmetically (matches jnp.pad 'reflect').
  const int tid = ty * 64 + tx;            // 0..255
  for (int idx = tid; idx < LH * LW; idx += 256) {
    const int i = idx / LW;
    const int j = idx - i * LW;
    int gr = by * TH - 1 + i;
    gr = (gr < 0) ? -gr : ((gr >= H) ? (2 * H - 2 - gr) : gr);
    int gc = bx * TW - 1 + j;
    gc = (gc < 0) ? -gc : ((gc >= W) ? (2 * W - 2 - gc) : gc);
    const size_t g = pbase + (size_t)gr * W + (size_t)gc;
#if USE_ASYNC_LDS
    __builtin_amdgcn_global_load_async_to_lds_b32(
        (g_int*)(pred + g), (l_int*)(xs + idx), 0, 0);
    __builtin_amdgcn_global_load_async_to_lds_b32(
        (g_int*)(targ + g), (l_int*)(ys + idx), 0, 0);
#else
    xs[idx] = pred[g];
    ys[idx] = targ[g];
#endif
  }
#if USE_ASYNC_LDS
  __builtin_amdgcn_s_wait_asynccnt(0);     // drain this wave's async copies
#endif
  __syncthreads();                          // make all waves' LDS data visible

  // ---------------- Separable 3x3 box sums via per-thread rolling window ----
  // Thread (tx,ty) produces tile rows ty*4 .. ty*4+3 at column tx.
  // For each staged row it computes the five horizontal 3-sums once and keeps
  // a 3-deep ring, so each output needs only the vertical reduction.
  const int base = ty * 4;                 // first staged row this thread touches
  constexpr float R9 = 1.0f / 9.0f;

  float h0x = 0.f, h1x = 0.f;
  float h0y = 0.f, h1y = 0.f;
  float h0xx = 0.f, h1xx = 0.f;
  float h0yy = 0.f, h1yy = 0.f;
  float h0xy = 0.f, h1xy = 0.f;

#pragma unroll
  for (int i = 0; i < 6; ++i) {
    const float* xr = xs + (base + i) * LW + tx;
    const float* yr = ys + (base + i) * LW + tx;
    const float x0 = xr[0], x1 = xr[1], x2 = xr[2];
    const float y0 = yr[0], y1 = yr[1], y2 = yr[2];

    const float nx  = x0 + x1 + x2;
    const float ny  = y0 + y1 + y2;
    const float nxx = x0 * x0 + x1 * x1 + x2 * x2;
    const float nyy = y0 * y0 + y1 * y1 + y2 * y2;
    const float nxy = x0 * y0 + x1 * y1 + x2 * y2;

    if (i >= 2) {
      const float Sx  = h0x  + h1x  + nx;
      const float Sy  = h0y  + h1y  + ny;
      const float Sxx = h0xx + h1xx + nxx;
      const float Syy = h0yy + h1yy + nyy;
      const float Sxy = h0xy + h1xy + nxy;

      const float mux   = Sx * R9;
      const float muy   = Sy * R9;
      const float sigx  = Sxx * R9 - mux * mux;
      const float sigy  = Syy * R9 - muy * muy;
      const float sigxy = Sxy * R9 - mux * muy;

      const float num = (2.0f * mux * muy + EPS1) * (2.0f * sigxy + EPS2);
      const float den = (mux * mux + muy * muy + EPS1) * (sigx + sigy + EPS2);
      float loss = (1.0f - num / den) * 0.5f;
      loss = fminf(fmaxf(loss, 0.0f), 1.0f);

      const int tr = base + i - 2;         // tile-local output row
      out[pbase + (size_t)(by * TH + tr) * W + (size_t)(bx * TW + tx)] = loss;
    }

    h0x = h1x;  h1x = nx;
    h0y = h1y;  h1y = ny;
    h0xx = h1xx; h1xx = nxx;
    h0yy = h1yy; h1yy = nyy;
    h0xy = h1xy; h1xy = nxy;
  }
}

extern "C" void kernel_launch(void* const* d_in, const int* in_sizes, int n_in,
                              void* d_out, int out_size, void* d_ws, size_t ws_size,
                              hipStream_t stream) {
  (void)n_in; (void)out_size; (void)d_ws; (void)ws_size;
  const float* pred = (const float*)d_in[0];
  const float* targ = (const float*)d_in[1];
  float* out = (float*)d_out;

  const int planes = in_sizes[0] / (H * W);       // 32*3 = 96
  dim3 grid(W / TW, H / TH, planes);              // 10 x 24 x 96
  dim3 block(TW, 4, 1);                           // 256 threads = 8 wave32
  ssim_err_kernel<<<grid, block, 0, stream>>>(pred, targ, out);
}